// SynapticPresyn_3229815406727
// MI455X (gfx1250) — compile-verified
//
#include <hip/hip_runtime.h>
#include <math.h>

typedef __attribute__((ext_vector_type(2))) float v2f;
typedef __attribute__((ext_vector_type(8))) float v8f;

#define Hh  12
#define Tt  2048
#define Dd  64
#define NHJ (Hh * Tt)          // 24576

#define EPSF      1e-6f
#define A_CA      0.25f
#define A_BON     0.6f
#define A_BOFF    0.1f
#define A_PRIME   0.1f
#define A_UNPRIME 0.04f
#define A_REFILL  0.04f
#define E_IN      0.03f
#define E_REL     0.015f
#define E_PUMP    0.006f
#define KD1       0.4f
#define KD7       3.0f
#define CPLX_BIAS 0.5f
#define QMAX      1.3f
#define QBETA     2.0f
#define BARRIER   0.075f
// exp(-1/tau) as double-rounded-to-float literals
#define RHO_C 0.7788007830714049f   // exp(-1/4)
#define RHO_B 0.9048374180359595f   // exp(-1/10)
#define RHO_P 0.9459594689067654f   // exp(-1/18)
#define RHO_R 0.9753099120283326f   // exp(-1/40)
#define RHO_E 0.9875778004938814f   // exp(-1/80)

__device__ __forceinline__ float sigm(float x) { return 1.0f / (1.0f + expf(-x)); }
__device__ __forceinline__ float clip01(float x) { return fminf(fmaxf(x, 0.0f), 1.0f); }

// ---------------- k1: partial column sums of softplus(clip(logits)) ----------------
// grid (8 j-blocks, 16 i-chunks of 128 rows, 12 heads), block 256
__global__ void k_influx_part(const float* __restrict__ logits, float* __restrict__ part) {
    int j  = blockIdx.x * 256 + threadIdx.x;
    int ic = blockIdx.y;
    int h  = blockIdx.z;
    const float* base = logits + (size_t)h * Tt * Tt;
    int i0 = ic * 128;
    float s = 0.0f;
    for (int r = 0; r < 128; ++r) {
        int i = i0 + r;
        float l = base[(size_t)i * Tt + j];
        l = fminf(fmaxf(l, -20.0f), 20.0f);
        float sp = log1pf(expf(l));
        s += (i >= j) ? sp : 0.0f;
    }
    part[((size_t)h * Tt + j) * 16 + ic] = s;
}

// ---------------- k2: reduce influx, state recurrences phase 1 ----------------
__global__ void k_state1(const float* __restrict__ part,
                         const float* __restrict__ C, const float* __restrict__ BUF,
                         const float* __restrict__ RRP, const float* __restrict__ RES,
                         const float* __restrict__ PR, const float* __restrict__ CL,
                         const float* __restrict__ E,
                         float* __restrict__ out_tail,
                         float* __restrict__ gate, float* __restrict__ avail,
                         float* __restrict__ prmid, float* __restrict__ resmid,
                         float* __restrict__ emid) {
    int idx = blockIdx.x * 256 + threadIdx.x;   // 0..NHJ-1
    int j = idx & (Tt - 1);
    const float* p = part + (size_t)idx * 16;
    float inf = 0.0f;
#pragma unroll
    for (int ic = 0; ic < 16; ++ic) inf += p[ic];
    inf /= (float)(Tt - j);                     // counts[j] = T - j >= 1

    float c0 = C[idx], b0 = BUF[idx];
    float c_new = fmaxf(RHO_C * c0 + A_CA * inf - A_BON * c0 * (1.0f - b0) + A_BOFF * b0, 0.0f);
    float b_new = clip01(RHO_B * b0 + A_BON * c0 * (1.0f - b0) - A_BOFF * b0);

    float pr_m  = clip01(RHO_P * PR[idx] + A_PRIME * (1.0f - PR[idx]));
    float rrp_r = clip01(RHO_R * RRP[idx] + A_REFILL * RES[idx]);
    float res_m = clip01(RES[idx] - A_REFILL * RES[idx]);
    float e_m   = fminf(fmaxf(RHO_E * E[idx] + E_IN, 0.0f), 1.6f);

    float syt = 0.7f * c_new / (c_new + KD1) + 0.3f * c_new / (c_new + KD7);
    float g = sigm(3.0f * syt + 2.0f * pr_m - 2.0f * (CL[idx] + CPLX_BIAS));

    out_tail[0 * NHJ + idx] = c_new;      // C_new
    out_tail[1 * NHJ + idx] = b_new;      // BUF_new
    out_tail[5 * NHJ + idx] = CL[idx];    // CL (pass-through)

    gate[idx] = g;
    avail[idx] = rrp_r;
    prmid[idx] = pr_m;
    resmid[idx] = res_m;
    emid[idx] = e_m;
}

// ---------------- k3: WMMA f32 K·Q^T + fused raw_release + column partials ----------------
// grid (128 j-tiles, 32 i-blocks of 64 rows, 12 heads), block 128 (4 waves)
__global__ void __launch_bounds__(128) k_wmma(const float* __restrict__ qm,
                                              const float* __restrict__ km,
                                              const float* __restrict__ gate,
                                              const float* __restrict__ avail,
                                              float* __restrict__ out,
                                              float* __restrict__ colpart) {
    __shared__ float cs[64];
    int lane = threadIdx.x & 31;
    int w    = threadIdx.x >> 5;
    int h    = blockIdx.z;
    int j0   = blockIdx.x << 4;
    int i0   = ((blockIdx.y << 2) | w) << 4;

    int mrow = lane & 15;
    int koff = (lane >> 4) << 1;      // 0 or 2: K pairs per half-wave (ISA f32 A/B layout)

    const float* kb = km + (size_t)h * Tt * Dd + (size_t)(i0 + mrow) * Dd + koff;
    const float* qb = qm + (size_t)h * Tt * Dd + (size_t)(j0 + mrow) * Dd + koff;

    v8f acc = {};
#pragma unroll
    for (int d0 = 0; d0 < Dd; d0 += 4) {
        v2f a = *(const v2f*)(kb + d0);   // A[M=mrow, K=d0+koff .. +1]
        v2f b = *(const v2f*)(qb + d0);   // B[K=d0+koff .. +1, N=mrow]
        acc = __builtin_amdgcn_wmma_f32_16x16x4_f32(false, a, false, b,
                                                    (short)0, acc, false, false);
    }

    // Epilogue: C/D layout — VGPR v: lanes 0-15 hold M=v, lanes 16-31 hold M=v+8; N = lane&15
    int ncol  = lane & 15;
    int j     = j0 + ncol;
    float g   = gate[h * Tt + j];
    float av  = avail[h * Tt + j];
    int mbase = (lane >> 4) << 3;
    float csum = 0.0f;
    float* orow = out + ((size_t)h * Tt + i0 + mbase) * Tt + j;
#pragma unroll
    for (int v = 0; v < 8; ++v) {
        int i = i0 + mbase + v;
        float s   = sigm(acc[v] * 0.125f);            // sigmoid(S / sqrt(64))
        float raw = (j <= i) ? clip01(g * s * av) : 0.0f;
        orow[(size_t)v * Tt] = raw;
        csum += raw;
    }
    // fold upper half-wave (M=8..15) into lower lanes, then reduce 4 waves via LDS
    float other = __shfl(csum, ncol + 16, 32);
    if (lane < 16) cs[w * 16 + ncol] = csum + other;
    __syncthreads();
    if (threadIdx.x < 16) {
        float t = cs[threadIdx.x] + cs[16 + threadIdx.x] + cs[32 + threadIdx.x] + cs[48 + threadIdx.x];
        colpart[((size_t)h * Tt + j0 + threadIdx.x) * 32 + blockIdx.y] = t;
    }
}

// ---------------- k4: per-column normalization + remaining states ----------------
__global__ void k_state2(const float* __restrict__ colpart,
                         const float* __restrict__ avail,
                         const float* __restrict__ prmid, const float* __restrict__ resmid,
                         const float* __restrict__ emid,
                         float* __restrict__ out_tail,
                         float* __restrict__ scale, float* __restrict__ lq) {
    int idx = blockIdx.x * 256 + threadIdx.x;   // 0..NHJ-1
    const float* p = colpart + (size_t)idx * 32;
    float tot = 0.0f;
#pragma unroll
    for (int r = 0; r < 32; ++r) tot += p[r];   // fixed order -> deterministic
    float av   = avail[idx];
    float sc   = fminf(1.0f, av / fmaxf(tot, EPSF));
    float used = tot * sc;
    float rrp  = clip01(av - used);
    float res  = clip01(resmid[idx] + used);
    float pr   = clip01(prmid[idx] - A_UNPRIME * used);
    float e    = fminf(fmaxf(emid[idx] - E_REL * used - E_PUMP * (1.0f - res), 0.0f), 1.6f);
    float qa   = sigm(QBETA * (e - 0.5f)) * QMAX;

    out_tail[2 * NHJ + idx] = rrp;   // RRP_new
    out_tail[3 * NHJ + idx] = res;   // RES_new
    out_tail[4 * NHJ + idx] = pr;    // PR_new
    out_tail[6 * NHJ + idx] = e;     // E_new
    scale[idx] = sc;
    lq[idx]    = fmaxf(qa, EPSF);
}

// ---------------- k5: in-place syn_logit ----------------
// grid (8, 2048 rows, 12 heads), block 256
__global__ void k_final(float* __restrict__ out,
                        const float* __restrict__ scale, const float* __restrict__ lq) {
    int j = blockIdx.x * 256 + threadIdx.x;
    int i = blockIdx.y;
    int h = blockIdx.z;
    size_t idx = ((size_t)h * Tt + i) * Tt + j;
    float raw = out[idx];
    float rf  = raw * scale[h * Tt + j];
    float v = logf(rf * lq[h * Tt + j] + EPSF)
            - BARRIER * fabsf((float)i - (float)j) * (1.0f / (float)Tt);
    out[idx] = v;
}

extern "C" void kernel_launch(void* const* d_in, const int* in_sizes, int n_in,
                              void* d_out, int out_size, void* d_ws, size_t ws_size,
                              hipStream_t stream) {
    const float* q      = (const float*)d_in[0];
    const float* k      = (const float*)d_in[1];
    const float* logits = (const float*)d_in[2];
    const float* C      = (const float*)d_in[3];
    const float* BUF    = (const float*)d_in[4];
    const float* RRP    = (const float*)d_in[5];
    const float* RES    = (const float*)d_in[6];
    const float* PR     = (const float*)d_in[7];
    const float* CL     = (const float*)d_in[8];
    const float* E      = (const float*)d_in[9];
    (void)in_sizes; (void)n_in; (void)out_size; (void)ws_size;

    float* out      = (float*)d_out;
    float* out_tail = out + (size_t)Hh * Tt * Tt;

    float* ws      = (float*)d_ws;
    float* part    = ws;                 // NHJ*16 floats
    float* gate    = ws + 16 * (size_t)NHJ;
    float* avail   = gate   + NHJ;
    float* prmid   = avail  + NHJ;
    float* resmid  = prmid  + NHJ;
    float* emid    = resmid + NHJ;
    float* scale   = emid   + NHJ;
    float* lq      = scale  + NHJ;
    float* colpart = lq     + NHJ;       // NHJ*32 floats  (total ~5.4 MB)

    k_influx_part<<<dim3(8, 16, Hh), 256, 0, stream>>>(logits, part);
    k_state1<<<NHJ / 256, 256, 0, stream>>>(part, C, BUF, RRP, RES, PR, CL, E,
                                            out_tail, gate, avail, prmid, resmid, emid);
    k_wmma<<<dim3(Tt / 16, Tt / 64, Hh), 128, 0, stream>>>(q, k, gate, avail, out, colpart);
    k_state2<<<NHJ / 256, 256, 0, stream>>>(colpart, avail, prmid, resmid, emid,
                                            out_tail, scale, lq);
    k_final<<<dim3(Tt / 256, Tt, Hh), 256, 0, stream>>>(out, scale, lq);
}